// SplashEncoding_52355651338845
// MI455X (gfx1250) — compile-verified
//
#include <hip/hip_runtime.h>

// SplashEncoding gather kernel for MI455X (gfx1250, wave32).
//   N = 524288 queries, G = 500000 gaussians, K = 8 neighbours, F = 32 features.
// Memory-bound gather: per-wave layout = 4 queries x 8 lanes,
//   Stage A: lane (q, k) computes gaussian weight for neighbour k of query q
//   Stage B: lane (q, k) accumulates features [4k..4k+3] of query q
// Neighbour indices are staged into LDS with the CDNA5 async gather engine
// (global_load_async_to_lds_b64 / ASYNCcnt) and {idx, weight} pairs are
// exchanged through LDS (intra-wave, in-order DS, no barrier needed).

#define KNN 8
#define FD  32

struct __align__(16) F4 { float x, y, z, w; };

__global__ __launch_bounds__(256) void splash_encoding_kernel(
    const float* __restrict__ coords,       // [N,3]
    const float* __restrict__ means,        // [G,3]
    const float* __restrict__ log_covs,     // [G,3]
    const float* __restrict__ feats,        // [G,32]
    const long long* __restrict__ nearest_idx, // [N,8]
    float* __restrict__ out,                // [N,32]
    int n)
{
    __shared__ unsigned long long sPair[8][32];   // 2 KB: per (wave, slot) {idx_lo, w_bits}

    const int tid  = threadIdx.x;
    const int wave = tid >> 5;
    const int lane = tid & 31;
    const int qi   = lane >> 3;   // query within wave (0..3)
    const int k    = lane & 7;    // neighbour id (stage A) / feature quad (stage B)

    const int q  = blockIdx.x * 32 + wave * 4 + qi;
    const int qc = (q < n) ? q : 0;

    // ---- Stage A: async-gather this wave's 32 int64 neighbour indices into LDS
    {
        unsigned ldsOff = (unsigned)(size_t)(&sPair[wave][lane]);
        unsigned long long ga =
            (unsigned long long)(const void*)(nearest_idx + (size_t)qc * KNN + k);
        asm volatile("global_load_async_to_lds_b64 %0, %1, off"
                     :: "v"(ldsOff), "v"(ga)
                     : "memory");
    }

    // coords load overlaps the async transfer (uniform across each 8-lane group)
    const float cx = coords[3 * (size_t)qc + 0];
    const float cy = coords[3 * (size_t)qc + 1];
    const float cz = coords[3 * (size_t)qc + 2];

    asm volatile("s_wait_asynccnt 0" ::: "memory");

    // low dword of the little-endian int64 index (values < G fit in 32 bits)
    int idx = ((const int*)&sPair[wave][lane])[0];
    int ic  = (idx < 0) ? 0 : idx;

    // gather mean / log_cov for (q, k); diagonal Mahalanobis -> gaussian weight
    const float mx = means[3 * (size_t)ic + 0];
    const float my = means[3 * (size_t)ic + 1];
    const float mz = means[3 * (size_t)ic + 2];
    const float lx = log_covs[3 * (size_t)ic + 0];
    const float ly = log_covs[3 * (size_t)ic + 1];
    const float lz = log_covs[3 * (size_t)ic + 2];

    const float dx = cx - mx, dy = cy - my, dz = cz - mz;
    const float md = dx * dx * __expf(-lx)
                   + dy * dy * __expf(-ly)
                   + dz * dz * __expf(-lz);
    float w = __expf(-0.5f * md);
    if (idx < 0) w = 0.0f;

    // publish {clamped idx, weight}; intra-wave DS ops stay in order
    sPair[wave][lane] =
        ((unsigned long long)__float_as_uint(w) << 32) | (unsigned)ic;

    // ---- Stage B: weighted feature sum; lane -> (query qi, features 4k..4k+3)
    const int slot = qi * 8;
    unsigned long long pr[KNN];
#pragma unroll
    for (int j = 0; j < KNN; ++j)
        pr[j] = sPair[wave][slot + j];

    // issue all 8 fully-coalesced 512B/wave gathers before consuming them
    F4 fv[KNN];
#pragma unroll
    for (int j = 0; j < KNN; ++j) {
        const size_t gi = (size_t)(unsigned)pr[j];
        fv[j] = *(const F4*)(feats + gi * FD + k * 4);
    }

    float ax = 0.f, ay = 0.f, az = 0.f, aw = 0.f;
#pragma unroll
    for (int j = 0; j < KNN; ++j) {
        const float wk = __uint_as_float((unsigned)(pr[j] >> 32));
        ax = fmaf(wk, fv[j].x, ax);
        ay = fmaf(wk, fv[j].y, ay);
        az = fmaf(wk, fv[j].z, az);
        aw = fmaf(wk, fv[j].w, aw);
    }

    if (q < n) {
        F4* o = (F4*)(out + (size_t)q * FD + k * 4);
        *o = F4{ax, ay, az, aw};
    }
}

extern "C" void kernel_launch(void* const* d_in, const int* in_sizes, int n_in,
                              void* d_out, int out_size, void* d_ws, size_t ws_size,
                              hipStream_t stream) {
    const float*     coords      = (const float*)d_in[0];
    const float*     means       = (const float*)d_in[1];
    const float*     log_covs    = (const float*)d_in[2];
    const float*     feats       = (const float*)d_in[3];
    const long long* nearest_idx = (const long long*)d_in[4];
    float*           out         = (float*)d_out;

    const int n = in_sizes[0] / 3;            // N = 524288
    const int blocks = (n + 31) / 32;         // 32 queries per 256-thread block

    splash_encoding_kernel<<<blocks, 256, 0, stream>>>(
        coords, means, log_covs, feats, nearest_idx, out, n);
}